// Interpolation_31645319037206
// MI455X (gfx1250) — compile-verified
//
#include <hip/hip_runtime.h>

typedef __attribute__((ext_vector_type(2))) float v2f;
typedef __attribute__((ext_vector_type(4))) float v4f;
typedef __attribute__((ext_vector_type(8))) float v8f;

#define NCOL 16

// NaN squash helpers (must happen BEFORE wmma: NaN propagates through it)
#define SQ(v) (((v) == (v)) ? (v) : 0.0f)
#define MK(v) (((v) == (v)) ? 1.0f : 0.0f)

// ---------------------------------------------------------------------------
// Pass 0: zero the 32-float accumulator workspace (sums[16], counts[16]).
// Must run every call: harness poisons d_ws once and never re-poisons.
// ---------------------------------------------------------------------------
__global__ void ws_init_kernel(float* __restrict__ ws) {
    int i = threadIdx.x;
    if (i < 2 * NCOL) ws[i] = 0.0f;
}

// ---------------------------------------------------------------------------
// Pass 1: masked column sum + valid count via V_WMMA_F32_16X16X4_F32.
//   A (16x4) = ones  ->  D[m,n] = sum_k B[k,n] + C[m,n]  (rows replicated)
//   B (4x16) = 4 time rows x 16 cols, NaN->0   => column sums
//   B'(4x16) = validity mask (1.0 / 0.0)       => column counts
// 16 rows (one "superchunk", 1KB) per iteration: 8 clause-able b32 loads,
// 8 WMMA across two accumulator pairs so dependent WMMAs sit 4 apart.
// Scalar (readfirstlane) loop control keeps the loop on SALU.
// EXEC is all-ones everywhere a WMMA executes.
// ---------------------------------------------------------------------------
__global__ __launch_bounds__(256) void nanmean_reduce_kernel(
        const float* __restrict__ x, float* __restrict__ ws, int T) {
    __shared__ float s_sum[NCOL];
    __shared__ float s_cnt[NCOL];
    if (threadIdx.x < NCOL) { s_sum[threadIdx.x] = 0.0f; s_cnt[threadIdx.x] = 0.0f; }
    __syncthreads();

    const int lane  = threadIdx.x & 31;
    const int col   = lane & 15;
    const int krow  = (lane >> 4) << 1;                 // 0 (lanes 0-15) or 2
    const int wpb   = blockDim.x >> 5;
    const int waveS = __builtin_amdgcn_readfirstlane(
                          (int)blockIdx.x * wpb + (int)(threadIdx.x >> 5));
    const int nwave = (int)gridDim.x * wpb;
    const int nSup  = T >> 4;                           // 16 rows per superchunk

    v8f s0 = {}, s1 = {};                               // column-sum accumulators
    v8f c0 = {}, c1 = {};                               // column-count accumulators
    const v2f ones = {1.0f, 1.0f};

    const float* p = x + krow * NCOL + col;             // per-lane base

    for (int sc = waveS; sc < nSup; sc += nwave) {
        const float* q = p + ((size_t)sc << 8);         // 256 floats / superchunk
        // Prefetch this wave's next superchunk (waves stride ~nwave KB apart).
        __builtin_prefetch(q + ((size_t)nwave << 8), 0, 1);

        float a0 = q[0],   a1 = q[16];                  // rows  krow,  krow+1
        float a2 = q[64],  a3 = q[80];                  // rows +4
        float a4 = q[128], a5 = q[144];                 // rows +8
        float a6 = q[192], a7 = q[208];                 // rows +12

        v2f bS, bM;
        bS = (v2f){SQ(a0), SQ(a1)};  bM = (v2f){MK(a0), MK(a1)};
        s0 = __builtin_amdgcn_wmma_f32_16x16x4_f32(false, ones, false, bS, (short)0, s0, false, false);
        c0 = __builtin_amdgcn_wmma_f32_16x16x4_f32(false, ones, false, bM, (short)0, c0, false, false);

        bS = (v2f){SQ(a2), SQ(a3)};  bM = (v2f){MK(a2), MK(a3)};
        s1 = __builtin_amdgcn_wmma_f32_16x16x4_f32(false, ones, false, bS, (short)0, s1, false, false);
        c1 = __builtin_amdgcn_wmma_f32_16x16x4_f32(false, ones, false, bM, (short)0, c1, false, false);

        bS = (v2f){SQ(a4), SQ(a5)};  bM = (v2f){MK(a4), MK(a5)};
        s0 = __builtin_amdgcn_wmma_f32_16x16x4_f32(false, ones, false, bS, (short)0, s0, false, false);
        c0 = __builtin_amdgcn_wmma_f32_16x16x4_f32(false, ones, false, bM, (short)0, c0, false, false);

        bS = (v2f){SQ(a6), SQ(a7)};  bM = (v2f){MK(a6), MK(a7)};
        s1 = __builtin_amdgcn_wmma_f32_16x16x4_f32(false, ones, false, bS, (short)0, s1, false, false);
        c1 = __builtin_amdgcn_wmma_f32_16x16x4_f32(false, ones, false, bM, (short)0, c1, false, false);
    }

    // D row M=0 = vector element 0, lanes 0-15 (N = lane). Rows identical;
    // use only lanes 0-15 to avoid double counting.
    float fs = s0[0] + s1[0];
    float fc = c0[0] + c1[0];
    if (lane < NCOL) {
        atomicAdd(&s_sum[lane], fs);                    // ds_add_f32
        atomicAdd(&s_cnt[lane], fc);
    }

    // Scalar tail for T % 16 rows (empty for T = 2,000,000; kept generic).
    if (blockIdx.x == 0 && threadIdx.x < NCOL) {
        float ts = 0.0f, tc = 0.0f;
        for (int r = nSup << 4; r < T; ++r) {
            float v = x[(size_t)r * NCOL + threadIdx.x];
            if (v == v) { ts += v; tc += 1.0f; }
        }
        if (ts != 0.0f || tc != 0.0f) {
            atomicAdd(&s_sum[threadIdx.x], ts);
            atomicAdd(&s_cnt[threadIdx.x], tc);
        }
    }
    __syncthreads();

    if (threadIdx.x < NCOL) {
        atomicAdd(&ws[threadIdx.x],        s_sum[threadIdx.x]);  // global_atomic_add_f32
        atomicAdd(&ws[NCOL + threadIdx.x], s_cnt[threadIdx.x]);
    }
}

// ---------------------------------------------------------------------------
// Pass 2: mean[c] = sum[c] / max(count[c], 1); out = isnan(x) ? mean : x.
// float4 streaming; non-temporal stores keep x's lines (hot in L2 from
// pass 1) resident so pass-2 reads hit L2 instead of HBM.
// ---------------------------------------------------------------------------
__global__ __launch_bounds__(256) void nanfill_kernel(
        const float* __restrict__ x, const float* __restrict__ ws,
        float* __restrict__ out, int n) {
    __shared__ float s_mean[NCOL];
    if (threadIdx.x < NCOL) {
        float s = ws[threadIdx.x];
        float c = ws[NCOL + threadIdx.x];
        s_mean[threadIdx.x] = s / fmaxf(c, 1.0f);
    }
    __syncthreads();

    const int n4     = n >> 2;
    const int stride = (int)gridDim.x * (int)blockDim.x;   // multiple of 4
    int i4 = (int)blockIdx.x * (int)blockDim.x + (int)threadIdx.x;
    const int base = (i4 << 2) & 15;                       // col of elem 0 (stride%4==0)

    #pragma unroll 2
    for (; i4 < n4; i4 += stride) {
        v4f v = ((const v4f*)x)[i4];
        v4f o;
        o.x = (v.x == v.x) ? v.x : s_mean[base + 0];
        o.y = (v.y == v.y) ? v.y : s_mean[base + 1];
        o.z = (v.z == v.z) ? v.z : s_mean[base + 2];
        o.w = (v.w == v.w) ? v.w : s_mean[base + 3];
        __builtin_nontemporal_store(o, (v4f*)out + i4);
    }
    // Tail elements (n % 4) — n = 32M here so empty, kept for safety.
    if (blockIdx.x == 0 && (int)threadIdx.x < (n & 3)) {
        int i = (n4 << 2) + (int)threadIdx.x;
        float v = x[i];
        out[i] = (v == v) ? v : s_mean[i & 15];
    }
}

extern "C" void kernel_launch(void* const* d_in, const int* in_sizes, int n_in,
                              void* d_out, int out_size, void* d_ws, size_t ws_size,
                              hipStream_t stream) {
    const float* x   = (const float*)d_in[0];
    float*       out = (float*)d_out;
    float*       ws  = (float*)d_ws;

    const int n = in_sizes[0];    // B*T*C = 32,000,000
    const int T = n / NCOL;       // 2,000,000

    ws_init_kernel<<<1, 64, 0, stream>>>(ws);

    // 512 blocks x 256 threads = 4096 waves; each wave handles ~30 superchunks
    // (488 rows) sequentially per column -> short fp32 accumulation chains.
    nanmean_reduce_kernel<<<512, 256, 0, stream>>>(x, ws, T);

    // Bandwidth-bound fill: float4 grid-stride with NT stores.
    nanfill_kernel<<<2048, 256, 0, stream>>>(x, ws, out, n);
}